// LightGCNLayer_46943992545845
// MI455X (gfx1250) — compile-verified
//
#include <hip/hip_runtime.h>
#include <stdint.h>

typedef int int4v __attribute__((ext_vector_type(4)));  // NT-load-compatible 128-bit vector

// ===================== common small kernels =====================

__global__ void zero_u32_kernel(uint32_t* __restrict__ p, int n) {
  int i = blockIdx.x * blockDim.x + threadIdx.x;
  int stride = gridDim.x * blockDim.x;
  for (int k = i; k < n; k += stride) p[k] = 0u;
}

__global__ void zero_f4_kernel(float4* __restrict__ p, int n4) {
  int i = blockIdx.x * blockDim.x + threadIdx.x;
  int stride = gridDim.x * blockDim.x;
  float4 z = make_float4(0.f, 0.f, 0.f, 0.f);
  for (int k = i; k < n4; k += stride) p[k] = z;
}

// degree count (u32 atomics, int4-vectorized non-temporal index loads)
__global__ void degree_kernel(const int* __restrict__ src, const int* __restrict__ dst,
                              int n4, int E, uint32_t* __restrict__ deg_src,
                              uint32_t* __restrict__ deg_dst) {
  const int4v* __restrict__ src4 = (const int4v*)src;
  const int4v* __restrict__ dst4 = (const int4v*)dst;
  int i = blockIdx.x * blockDim.x + threadIdx.x;
  int stride = gridDim.x * blockDim.x;
  for (int k = i; k < n4; k += stride) {
    int4v s = __builtin_nontemporal_load(&src4[k]);
    atomicAdd(&deg_src[s.x], 1u); atomicAdd(&deg_src[s.y], 1u);
    atomicAdd(&deg_src[s.z], 1u); atomicAdd(&deg_src[s.w], 1u);
    int4v d = __builtin_nontemporal_load(&dst4[k]);
    atomicAdd(&deg_dst[d.x], 1u); atomicAdd(&deg_dst[d.y], 1u);
    atomicAdd(&deg_dst[d.z], 1u); atomicAdd(&deg_dst[d.w], 1u);
  }
  for (int e = n4 * 4 + i; e < E; e += stride) {
    atomicAdd(&deg_src[src[e]], 1u);
    atomicAdd(&deg_dst[dst[e]], 1u);
  }
}

// deg -> 1/max(deg,1), in place (u32 slot becomes f32)
__global__ void norm_kernel(uint32_t* __restrict__ deg, int n) {
  int i = blockIdx.x * blockDim.x + threadIdx.x;
  int stride = gridDim.x * blockDim.x;
  for (int k = i; k < n; k += stride) {
    uint32_t c = deg[k];
    float f = 1.0f / (float)(c > 1u ? c : 1u);
    ((float*)deg)[k] = f;
  }
}

// ===================== CSR build: 3-kernel exclusive scan =====================
__global__ __launch_bounds__(256) void scan_block_kernel(
    const uint32_t* __restrict__ counts, int n,
    uint32_t* __restrict__ start, uint32_t* __restrict__ block_sums) {
  __shared__ uint32_t lds[256];
  const int t = threadIdx.x;
  const int i0 = blockIdx.x * 1024 + t * 4;
  uint32_t v0 = (i0 + 0 < n) ? counts[i0 + 0] : 0u;
  uint32_t v1 = (i0 + 1 < n) ? counts[i0 + 1] : 0u;
  uint32_t v2 = (i0 + 2 < n) ? counts[i0 + 2] : 0u;
  uint32_t v3 = (i0 + 3 < n) ? counts[i0 + 3] : 0u;
  uint32_t s1 = v0, s2 = s1 + v1, s3 = s2 + v2, s4 = s3 + v3;
  lds[t] = s4;
  __syncthreads();
#pragma unroll
  for (int off = 1; off < 256; off <<= 1) {
    uint32_t x = (t >= off) ? lds[t - off] : 0u;
    __syncthreads();
    lds[t] += x;
    __syncthreads();
  }
  uint32_t prev = (t > 0) ? lds[t - 1] : 0u;
  if (i0 + 0 < n) start[i0 + 1] = prev + s1;
  if (i0 + 1 < n) start[i0 + 2] = prev + s2;
  if (i0 + 2 < n) start[i0 + 3] = prev + s3;
  if (i0 + 3 < n) start[i0 + 4] = prev + s4;
  if (t == 255) block_sums[blockIdx.x] = lds[255];
}

__global__ __launch_bounds__(256) void scan_sums_kernel(uint32_t* __restrict__ sums, int nb) {
  __shared__ uint32_t lds[256];
  const int t = threadIdx.x;
  uint32_t v = (t < nb) ? sums[t] : 0u;
  lds[t] = v;
  __syncthreads();
#pragma unroll
  for (int off = 1; off < 256; off <<= 1) {
    uint32_t x = (t >= off) ? lds[t - off] : 0u;
    __syncthreads();
    lds[t] += x;
    __syncthreads();
  }
  if (t < nb) sums[t] = lds[t];
}

__global__ void scan_finalize_kernel(uint32_t* __restrict__ start,
                                     const uint32_t* __restrict__ sums_incl,
                                     uint32_t* __restrict__ counts_to_cursor, int n) {
  int i = blockIdx.x * blockDim.x + threadIdx.x;
  if (i == 0) start[0] = 0u;
  if (i < n) {
    int b = i >> 10;
    uint32_t off = (b > 0) ? sums_incl[b - 1] : 0u;
    uint32_t incl = start[i + 1] + off;
    start[i + 1] = incl;
    uint32_t c = counts_to_cursor[i];
    counts_to_cursor[i] = incl - c;  // exclusive start -> cursor init
  }
}

// bin edges by destination: 1.6M returning u32 atomics (vs 102M f32 atomics push-style)
__global__ void reorder_kernel(const int* __restrict__ src, const int* __restrict__ dst,
                               int E, uint32_t* __restrict__ cursor,
                               uint32_t* __restrict__ esrc) {
  int i = blockIdx.x * blockDim.x + threadIdx.x;
  int stride = gridDim.x * blockDim.x;
  for (int e = i; e < E; e += stride) {
    int d = __builtin_nontemporal_load(&dst[e]);
    int s = __builtin_nontemporal_load(&src[e]);
    uint32_t p = atomicAdd(&cursor[d], 1u);
    esrc[p] = (uint32_t)s;
  }
}

// optional: feat = h * norm_src (removes per-edge norm lookups in gather)
__global__ void prescale_kernel(const float4* __restrict__ h4,
                                const float* __restrict__ norm_src,
                                float4* __restrict__ feat4, int n4) {
  int i = blockIdx.x * blockDim.x + threadIdx.x;
  int stride = gridDim.x * blockDim.x;
  for (int k = i; k < n4; k += stride) {
    float f = norm_src[k >> 4];
    float4 v = h4[k];
    v.x *= f; v.y *= f; v.z *= f; v.w *= f;
    feat4[k] = v;
  }
}

// ===================== pull-gather: wave32 per destination row =====================
// Lanes batch-load up to 32 edge ids (NT stream), v_readlane broadcasts each source id,
// 2x-unrolled register accumulation (two b64 loads in flight), one 256B row store.
template <bool PRESCALED>
__global__ __launch_bounds__(256) void gather_kernel(
    const float* __restrict__ feat, const uint32_t* __restrict__ esrc,
    const uint32_t* __restrict__ start, const float* __restrict__ norm_src,
    float* __restrict__ out, int n_dst) {
  const int lane = (int)(threadIdx.x & 31u);
  const int wave =
      __builtin_amdgcn_readfirstlane((int)((blockIdx.x * blockDim.x + threadIdx.x) >> 5));
  const int nwaves = (int)((gridDim.x * blockDim.x) >> 5);

  for (int d = wave; d < n_dst; d += nwaves) {
    uint32_t s0 = start[d];
    uint32_t s1 = start[d + 1];
    int cnt = (int)(s1 - s0);
    float a0 = 0.f, a1 = 0.f;  // even-j accumulators
    float b0 = 0.f, b1 = 0.f;  // odd-j accumulators

    for (int b = 0; b < cnt; b += 32) {
      int m = cnt - b;
      if (m > 32) m = 32;
      int s = 0;
      if (lane < m) {
        s = (int)__builtin_nontemporal_load(&esrc[(size_t)s0 + b + lane]);
        // warm both 128B lines of this lane's source row (global_prefetch_b8)
        __builtin_prefetch(&feat[(size_t)s * 64 + lane], 0, 0);
        __builtin_prefetch(&feat[(size_t)s * 64 + 32 + lane], 0, 0);
      }
      int j = 0;
      for (; j + 1 < m; j += 2) {
        int ss0 = __builtin_amdgcn_readlane(s, j);
        int ss1 = __builtin_amdgcn_readlane(s, j + 1);
        const float2 v0 = *(const float2*)(feat + (size_t)ss0 * 64 + lane * 2);
        const float2 v1 = *(const float2*)(feat + (size_t)ss1 * 64 + lane * 2);
        if (PRESCALED) {
          a0 += v0.x; a1 += v0.y;
          b0 += v1.x; b1 += v1.y;
        } else {
          float n0 = norm_src[ss0];
          float n1 = norm_src[ss1];
          a0 += v0.x * n0; a1 += v0.y * n0;
          b0 += v1.x * n1; b1 += v1.y * n1;
        }
      }
      if (j < m) {
        int ss = __builtin_amdgcn_readlane(s, j);
        const float2 v = *(const float2*)(feat + (size_t)ss * 64 + lane * 2);
        if (PRESCALED) {
          a0 += v.x; a1 += v.y;
        } else {
          float ns = norm_src[ss];
          a0 += v.x * ns; a1 += v.y * ns;
        }
      }
    }
    float nd = 1.0f / (float)(cnt > 1 ? cnt : 1);
    float2 r;
    r.x = (a0 + b0) * nd;
    r.y = (a1 + b1) * nd;
    *(float2*)(out + (size_t)d * 64 + lane * 2) = r;  // single row store, full overwrite
  }
}

// ===================== fallback: push-scatter with f32 atomics =====================
__global__ __launch_bounds__(256) void scatter_kernel(
    const float* __restrict__ h, const int* __restrict__ src, const int* __restrict__ dst,
    const float* __restrict__ norm_src, float* __restrict__ out, int E) {
  const int lane = (int)(threadIdx.x & 31u);
  const int wave = (int)((blockIdx.x * blockDim.x + threadIdx.x) >> 5);
  const int nwaves = (int)((gridDim.x * blockDim.x) >> 5);
  for (int base = wave * 32; base < E; base += nwaves * 32) {
    int nb = base + nwaves * 32;
    if (nb < E) {
      __builtin_prefetch(&src[nb + lane], 0, 0);
      __builtin_prefetch(&dst[nb + lane], 0, 0);
    }
    int e = base + lane;
    int s = 0, d = 0;
    if (e < E) { s = src[e]; d = dst[e]; }
    int cnt = E - base;
    int m = cnt < 32 ? cnt : 32;
    for (int j = 0; j < m; ++j) {
      int ss = __builtin_amdgcn_readlane(s, j);
      int dd = __builtin_amdgcn_readlane(d, j);
      float ns = norm_src[ss];
      const float2 v = *(const float2*)(h + (size_t)ss * 64 + lane * 2);
      float* o = out + (size_t)dd * 64 + lane * 2;
      unsafeAtomicAdd(o, v.x * ns);
      unsafeAtomicAdd(o + 1, v.y * ns);
    }
  }
}

__global__ void scale_kernel(float4* __restrict__ out4, const float* __restrict__ norm_dst,
                             int n4) {
  int i = blockIdx.x * blockDim.x + threadIdx.x;
  int stride = gridDim.x * blockDim.x;
  for (int k = i; k < n4; k += stride) {
    float f = norm_dst[k >> 4];
    float4 v = out4[k];
    v.x *= f; v.y *= f; v.z *= f; v.w *= f;
    out4[k] = v;
  }
}

// ===================== host orchestration =====================
extern "C" void kernel_launch(void* const* d_in, const int* in_sizes, int n_in,
                              void* d_out, int out_size, void* d_ws, size_t ws_size,
                              hipStream_t stream) {
  const float* h   = (const float*)d_in[0];
  const int*   src = (const int*)d_in[1];
  const int*   dst = (const int*)d_in[2];
  const int N_SRC = in_sizes[0] / 64;
  const int E     = in_sizes[1];
  const int N_DST = out_size / 64;
  float* out = (float*)d_out;

  // workspace carve-out (256B aligned slices)
  char* ws = (char*)d_ws;
  size_t off = 0;
  auto carve = [&](size_t bytes) -> void* {
    void* p = ws + off;
    off += (bytes + 255) & ~(size_t)255;
    return p;
  };
  uint32_t* deg_src = (uint32_t*)carve((size_t)N_SRC * 4);
  uint32_t* deg_dst = (uint32_t*)carve((size_t)N_DST * 4);
  size_t deg_words = ((size_t)((char*)deg_dst - ws)) / 4 + (size_t)N_DST;  // zero prefix span
  uint32_t* start   = (uint32_t*)carve(((size_t)N_DST + 1) * 4);
  int nb_scan = (N_DST + 1023) / 1024;
  uint32_t* bsums   = (uint32_t*)carve((size_t)nb_scan * 4);
  uint32_t* esrc    = (uint32_t*)carve((size_t)E * 4);
  size_t need_csr = off;
  float* feat = (float*)carve((size_t)N_SRC * 64 * 4);
  size_t need_feat = off;

  const bool can_csr  = (ws_size >= need_csr) && (nb_scan <= 256);
  const bool can_feat = can_csr && (ws_size >= need_feat);

  // 1) zero degree counters (ws is poisoned by harness)
  zero_u32_kernel<<<256, 256, 0, stream>>>((uint32_t*)ws, (int)deg_words);

  // 2) degrees
  degree_kernel<<<1536, 256, 0, stream>>>(src, dst, E / 4, E, deg_src, deg_dst);

  if (can_csr) {
    // 3) norm_src in place (deg_dst stays integer for the scan)
    norm_kernel<<<(N_SRC + 255) / 256, 256, 0, stream>>>(deg_src, N_SRC);

    // 4) exclusive scan of in-degrees -> start[], cursor in deg_dst
    scan_block_kernel<<<nb_scan, 256, 0, stream>>>(deg_dst, N_DST, start, bsums);
    scan_sums_kernel<<<1, 256, 0, stream>>>(bsums, nb_scan);
    scan_finalize_kernel<<<(N_DST + 255) / 256, 256, 0, stream>>>(start, bsums, deg_dst, N_DST);

    // 5) bin edges by destination
    reorder_kernel<<<2048, 256, 0, stream>>>(src, dst, E, deg_dst, esrc);

    // 6) gather (optionally from pre-scaled features)
    if (can_feat) {
      prescale_kernel<<<2048, 256, 0, stream>>>((const float4*)h, (const float*)deg_src,
                                                (float4*)feat, N_SRC * 16);
      gather_kernel<true><<<2048, 256, 0, stream>>>(feat, esrc, start, (const float*)deg_src,
                                                    out, N_DST);
    } else {
      gather_kernel<false><<<2048, 256, 0, stream>>>(h, esrc, start, (const float*)deg_src,
                                                     out, N_DST);
    }
  } else {
    // -------- fallback: atomic push-scatter (needs only ~800KB of ws) --------
    zero_f4_kernel<<<2048, 256, 0, stream>>>((float4*)d_out, out_size / 4);
    norm_kernel<<<(N_SRC + 255) / 256, 256, 0, stream>>>(deg_src, N_SRC);
    norm_kernel<<<(N_DST + 255) / 256, 256, 0, stream>>>(deg_dst, N_DST);
    scatter_kernel<<<1280, 256, 0, stream>>>(h, src, dst, (const float*)deg_src, out, E);
    scale_kernel<<<1024, 256, 0, stream>>>((float4*)d_out, (const float*)deg_dst, out_size / 4);
  }
}